// trans_IGN_5136780886777
// MI455X (gfx1250) — compile-verified
//
#include <hip/hip_runtime.h>
#include <hip/hip_bf16.h>

typedef __attribute__((ext_vector_type(2))) float v2f;
typedef __attribute__((ext_vector_type(8))) float v8f;

#define NEG_INF_ORD 0x007FFFFFu  // ordered-uint encoding of -inf

__device__ __forceinline__ float lrelu_f(float x) { return x > 0.0f ? x : 0.01f * x; }

// Monotonic float <-> uint mapping so atomicMax(unsigned) implements float max.
__device__ __forceinline__ unsigned f2ord(float f) {
    unsigned u = __float_as_uint(f);
    return (u & 0x80000000u) ? ~u : (u | 0x80000000u);
}
__device__ __forceinline__ float ord2f(unsigned u) {
    return (u & 0x80000000u) ? __uint_as_float(u & 0x7fffffffu) : __uint_as_float(~u);
}

#define WMMA_F32(a, b, c) \
    __builtin_amdgcn_wmma_f32_16x16x4_f32(false, (a), false, (b), (short)0, (c), false, false)

// ---------------------------------------------------------------------------
// WMMA fp32 GEMM:  C[M,Nn] = act(A[M,lda] @ W[Nn,ldw]^T + bias[Nn])
// REQUIRES: K % 4 == 0, lda % 2 == 0, ldw % 2 == 0 (call sites pad to this).
// Block = 128 threads (4 waves). Wave w owns rows [bx*128 + 32w, +32) as two
// 16-row tiles x four 16-col tiles (8 accumulators). Hot loop: 6 aligned
// b64 loads + 8 WMMAs, zero branches. act: 0=none, 1=leaky_relu(0.01).
// ---------------------------------------------------------------------------
__global__ __launch_bounds__(128) void gemm_wmma_f32(
    const float* __restrict__ A, int lda,
    const float* __restrict__ W, int ldw,
    const float* __restrict__ bias,
    float* __restrict__ C, int ldc,
    int M, int Nn, int K, int act)
{
    const int lane = threadIdx.x & 31;
    const int wave = threadIdx.x >> 5;
    const int half = lane >> 4;   // 0: K 0..1 / M 0..7    1: K 2..3 / M 8..15
    const int l16  = lane & 15;
    const int row0 = blockIdx.x * 128 + wave * 32;
    const int col0 = blockIdx.y * 64;

    // Clamped (always in-bounds) base pointers; out-of-range rows/cols are
    // computed on duplicate data and masked at the store.
    const int m0 = min(row0 + l16,      M - 1);
    const int m1 = min(row0 + 16 + l16, M - 1);
    const float* pa0 = A + (size_t)m0 * lda + half * 2;
    const float* pa1 = A + (size_t)m1 * lda + half * 2;
    const float* pb0 = W + (size_t)min(col0 +  0 + l16, Nn - 1) * ldw + half * 2;
    const float* pb1 = W + (size_t)min(col0 + 16 + l16, Nn - 1) * ldw + half * 2;
    const float* pb2 = W + (size_t)min(col0 + 32 + l16, Nn - 1) * ldw + half * 2;
    const float* pb3 = W + (size_t)min(col0 + 48 + l16, Nn - 1) * ldw + half * 2;

    v8f c00 = {}, c01 = {}, c02 = {}, c03 = {};
    v8f c10 = {}, c11 = {}, c12 = {}, c13 = {};

    for (int k0 = 0; k0 < K; k0 += 4) {
        const v2f a0 = *(const v2f*)(pa0 + k0);
        const v2f a1 = *(const v2f*)(pa1 + k0);
        const v2f b0 = *(const v2f*)(pb0 + k0);
        const v2f b1 = *(const v2f*)(pb1 + k0);
        const v2f b2 = *(const v2f*)(pb2 + k0);
        const v2f b3 = *(const v2f*)(pb3 + k0);
        c00 = WMMA_F32(a0, b0, c00);
        c01 = WMMA_F32(a0, b1, c01);
        c02 = WMMA_F32(a0, b2, c02);
        c03 = WMMA_F32(a0, b3, c03);
        c10 = WMMA_F32(a1, b0, c10);
        c11 = WMMA_F32(a1, b1, c11);
        c12 = WMMA_F32(a1, b2, c12);
        c13 = WMMA_F32(a1, b3, c13);
    }

#define STORE_TILE(accv, rbase, cbase)                                          \
    {                                                                           \
        const int n = (cbase) + l16;                                            \
        if (n < Nn) {                                                           \
            const float bv = bias ? bias[n] : 0.0f;                             \
            _Pragma("unroll")                                                   \
            for (int v = 0; v < 8; ++v) {                                       \
                const int r = (rbase) + v + half * 8;                           \
                if (r < M) {                                                    \
                    float val = (accv)[v] + bv;                                 \
                    if (act == 1) val = lrelu_f(val);                           \
                    C[(size_t)r * ldc + n] = val;                               \
                }                                                               \
            }                                                                   \
        }                                                                       \
    }

    STORE_TILE(c00, row0,      col0 +  0)
    STORE_TILE(c01, row0,      col0 + 16)
    STORE_TILE(c02, row0,      col0 + 32)
    STORE_TILE(c03, row0,      col0 + 48)
    STORE_TILE(c10, row0 + 16, col0 +  0)
    STORE_TILE(c11, row0 + 16, col0 + 16)
    STORE_TILE(c12, row0 + 16, col0 + 32)
    STORE_TILE(c13, row0 + 16, col0 + 48)
#undef STORE_TILE
}

// ---------------------------------------------------------------------------
// Elementwise / glue kernels
// ---------------------------------------------------------------------------
__global__ void fill_f32_k(float* p, float v, size_t n) {
    size_t i = (size_t)blockIdx.x * blockDim.x + threadIdx.x;
    if (i < n) p[i] = v;
}
__global__ void fill_u32_k(unsigned* p, unsigned v, size_t n) {
    size_t i = (size_t)blockIdx.x * blockDim.x + threadIdx.x;
    if (i < n) p[i] = v;
}

// Zero-pad a [rows, Ksrc] row-major matrix into [rows, Kdst]
__global__ void pad_w_k(float* __restrict__ dst, const float* __restrict__ src,
                        int rows, int Ksrc, int Kdst) {
    size_t i = (size_t)blockIdx.x * blockDim.x + threadIdx.x;
    if (i >= (size_t)rows * Kdst) return;
    int r = (int)(i / Kdst), c = (int)(i % Kdst);
    dst[i] = (c < Ksrc) ? src[(size_t)r * Ksrc + c] : 0.0f;
}

// X1[e, 0:40] = node[src[e]], X1[e, 40:50] = edge_feats[e], pad to ld
__global__ void build_X1_k(const float* __restrict__ node, const float* __restrict__ ef,
                           const int* __restrict__ src, float* __restrict__ X1,
                           int E, int NF, int EF, int ld) {
    size_t i = (size_t)blockIdx.x * blockDim.x + threadIdx.x;
    if (i >= (size_t)E * ld) return;
    int e = (int)(i / ld), j = (int)(i % ld);
    float v = 0.0f;
    if (j < NF)           v = node[(size_t)src[e] * NF + j];
    else if (j < NF + EF) v = ef[(size_t)e * EF + (j - NF)];
    X1[i] = v;
}

// X3[e, 0] = bond[e], X3[e, 1+g] = sumf[src[e]][g] + sumf[dst[e]][g], pad to ld
__global__ void build_X3_k(const float* __restrict__ bond, const float* __restrict__ sumf,
                           const int* __restrict__ src, const int* __restrict__ dst,
                           float* __restrict__ X3, int E, int G, int ld) {
    size_t i = (size_t)blockIdx.x * blockDim.x + threadIdx.x;
    if (i >= (size_t)E * ld) return;
    int e = (int)(i / ld), j = (int)(i % ld);
    float v = 0.0f;
    if (j == 0)          v = bond[e];
    else if (j <= G)     v = sumf[(size_t)src[e] * G + (j - 1)] +
                             sumf[(size_t)dst[e] * G + (j - 1)];
    X3[i] = v;
}

// logit[e] = lrelu(b + dot(W[0:G], Xd[dst[e]]) + dot(W[G:2G], Xb[srcIdx? srcIdx[e] : e]))
__global__ void edge_logits_k(const float* __restrict__ W, const float* __restrict__ b,
                              const float* __restrict__ Xd, const int* __restrict__ dst,
                              const float* __restrict__ Xb, const int* __restrict__ srcIdx,
                              float* __restrict__ out, int E, int G) {
    int e = blockIdx.x * blockDim.x + threadIdx.x;
    if (e >= E) return;
    const float* xd = Xd + (size_t)dst[e] * G;
    const float* xb = Xb + (size_t)(srcIdx ? srcIdx[e] : e) * G;
    float s = b[0];
    for (int g = 0; g < G; ++g) s += W[g] * xd[g] + W[G + g] * xb[g];
    out[e] = lrelu_f(s);
}

__global__ void segmax_k(const float* __restrict__ x, const int* __restrict__ idx,
                         unsigned* __restrict__ nmax, int E) {
    int e = blockIdx.x * blockDim.x + threadIdx.x;
    if (e >= E) return;
    atomicMax(&nmax[idx[e]], f2ord(x[e]));
}

__global__ void seg_expsum_k(float* __restrict__ x, const int* __restrict__ idx,
                             const unsigned* __restrict__ nmax, float* __restrict__ nsum, int E) {
    int e = blockIdx.x * blockDim.x + threadIdx.x;
    if (e >= E) return;
    float ex = expf(x[e] - ord2f(nmax[idx[e]]));
    x[e] = ex;
    atomicAdd(&nsum[idx[e]], ex);
}

// out[dst[e]][g] += (ex[e]/nsum[dst[e]] + addc) * V[(gsrc? gsrc[e] : e)][g]
__global__ void scatter_attn_k(const float* __restrict__ ex, const float* __restrict__ nsum,
                               const int* __restrict__ dst, const int* __restrict__ gsrc,
                               const float* __restrict__ V, float* __restrict__ out,
                               size_t total, int G, float addc) {
    size_t i = (size_t)blockIdx.x * blockDim.x + threadIdx.x;
    if (i >= total) return;
    int e = (int)(i / G), g = (int)(i % G);
    int d = dst[e];
    float coeff = ex[e] / nsum[d] + addc;
    float v = V[(size_t)(gsrc ? gsrc[e] : e) * G + g];
    atomicAdd(&out[(size_t)d * G + g], coeff * v);
}

__global__ void elu_ip_k(float* x, size_t n) {
    size_t i = (size_t)blockIdx.x * blockDim.x + threadIdx.x;
    if (i >= n) return;
    float v = x[i];
    x[i] = v > 0.0f ? v : expf(v) - 1.0f;
}

// torch-GRUCell math + ReLU: gi/gh are [Nrows, 3G] (r,z,n), hprev [Nrows, G]
__global__ void gru_relu_k(const float* __restrict__ gi, const float* __restrict__ gh,
                           const float* __restrict__ hprev, float* __restrict__ hout,
                           int Nrows, int G) {
    size_t i = (size_t)blockIdx.x * blockDim.x + threadIdx.x;
    if (i >= (size_t)Nrows * G) return;
    int r = (int)(i / G), c = (int)(i % G);
    size_t b0 = (size_t)r * 3 * G;
    float ir = gi[b0 + c], iz = gi[b0 + G + c], inn = gi[b0 + 2 * G + c];
    float hr = gh[b0 + c], hz = gh[b0 + G + c], hn  = gh[b0 + 2 * G + c];
    float rr = 1.0f / (1.0f + expf(-(ir + hr)));
    float zz = 1.0f / (1.0f + expf(-(iz + hz)));
    float nn = tanhf(inn + rr * hn);
    float h  = (1.0f - zz) * nn + zz * hprev[i];
    hout[i] = h > 0.0f ? h : 0.0f;
}

// Per-column mean/biased-var over M rows (block per column, LDS reduce)
__global__ __launch_bounds__(256) void colstats_k(const float* __restrict__ X, int M, int G,
                                                  float* __restrict__ mu, float* __restrict__ var) {
    __shared__ float ss[256], ss2[256];
    int c = blockIdx.x;
    float s = 0.0f, s2 = 0.0f;
    for (int r = threadIdx.x; r < M; r += 256) {
        float v = X[(size_t)r * G + c];
        s += v; s2 += v * v;
    }
    ss[threadIdx.x] = s; ss2[threadIdx.x] = s2;
    __syncthreads();
    for (int off = 128; off > 0; off >>= 1) {
        if (threadIdx.x < off) {
            ss[threadIdx.x]  += ss[threadIdx.x + off];
            ss2[threadIdx.x] += ss2[threadIdx.x + off];
        }
        __syncthreads();
    }
    if (threadIdx.x == 0) {
        float m = ss[0] / (float)M;
        mu[c]  = m;
        var[c] = ss2[0] / (float)M - m * m;
    }
}

// Y = (add? add : 0) + g*(X-mu)*rsqrt(var+1e-5) + b
__global__ void bn_apply_k(const float* __restrict__ X, const float* __restrict__ add,
                           const float* __restrict__ mu, const float* __restrict__ var,
                           const float* __restrict__ g, const float* __restrict__ b,
                           float* __restrict__ Y, size_t total, int G) {
    size_t i = (size_t)blockIdx.x * blockDim.x + threadIdx.x;
    if (i >= total) return;
    int c = (int)(i % G);
    float y = g[c] * (X[i] - mu[c]) * rsqrtf(var[c] + 1e-5f) + b[c];
    Y[i] = (add ? add[i] : 0.0f) + y;
}

// out[e] = act(b + dot(w, X[e,:K]));  act: 1=lrelu, 2=tanh
__global__ void rowdot_k(const float* __restrict__ X, int ld, const float* __restrict__ w,
                         const float* __restrict__ b, float* __restrict__ out,
                         int M, int K, int act) {
    int e = blockIdx.x * blockDim.x + threadIdx.x;
    if (e >= M) return;
    float s = b[0];
    for (int k = 0; k < K; ++k) s += w[k] * X[(size_t)e * ld + k];
    if (act == 1) s = lrelu_f(s);
    else if (act == 2) s = tanhf(s);
    out[e] = s;
}

__global__ void readout_scatter_k(const float* __restrict__ e3, const float* __restrict__ w,
                                  const int* __restrict__ egid, float* __restrict__ gsum,
                                  unsigned* __restrict__ gmax, int E, int G) {
    size_t i = (size_t)blockIdx.x * blockDim.x + threadIdx.x;
    if (i >= (size_t)E * G) return;
    int e = (int)(i / G), g = (int)(i % G);
    int gid = egid[e];
    float v = e3[i];
    atomicAdd(&gsum[(size_t)gid * G + g], w[e] * v);
    atomicMax(&gmax[(size_t)gid * G + g], f2ord(v));
}

__global__ void pack_out_k(const float* __restrict__ gsum, const unsigned* __restrict__ gmax,
                           float* __restrict__ out, int NG, int G) {
    size_t i = (size_t)blockIdx.x * blockDim.x + threadIdx.x;
    if (i >= (size_t)NG * G) return;
    int gid = (int)(i / G), g = (int)(i % G);
    out[(size_t)gid * 2 * G + g]     = gsum[i];
    out[(size_t)gid * 2 * G + G + g] = ord2f(gmax[i]);
}

// ---------------------------------------------------------------------------
// Host: wire the full pipeline
// ---------------------------------------------------------------------------
extern "C" void kernel_launch(void* const* d_in, const int* in_sizes, int n_in,
                              void* d_out, int out_size, void* d_ws, size_t ws_size,
                              hipStream_t stream) {
    (void)in_sizes; (void)n_in; (void)out_size; (void)ws_size;
    const int N = 50000, E = 200000, NG = 256, NF = 40, EF = 10, G = 200;
    const int KP1 = 52;   // padded K for he1 GEMM (50 -> 52)
    const int KP3 = 204;  // padded K for g3 layer-1 GEMM (201 -> 204)

    const float* node  = (const float*)d_in[0];
    const float* edgef = (const float*)d_in[1];
    const float* bond  = (const float*)d_in[2];
    const int*   src   = (const int*)d_in[3];
    const int*   dst   = (const int*)d_in[4];
    const int*   egid  = (const int*)d_in[5];

    int pi = 6;
    const float* gc_pn_W   = (const float*)d_in[pi++];
    const float* gc_pn_b   = (const float*)d_in[pi++];
    const float* gc_pe1_W  = (const float*)d_in[pi++];
    const float* gc_pe1_b  = (const float*)d_in[pi++];
    const float* gc_pe2_W  = (const float*)d_in[pi++];
    const float* gc_pe2_b  = (const float*)d_in[pi++];
    const float* gc_et_W   = (const float*)d_in[pi++];
    const float* gc_et_b   = (const float*)d_in[pi++];
    const float* gc_gru_Wih = (const float*)d_in[pi++];
    const float* gc_gru_Whh = (const float*)d_in[pi++];
    const float* gc_gru_bih = (const float*)d_in[pi++];
    const float* gc_gru_bhh = (const float*)d_in[pi++];
    const float* gl_pe_W   = (const float*)d_in[pi++];
    const float* gl_pe_b   = (const float*)d_in[pi++];
    const float* gl_pn_W   = (const float*)d_in[pi++];
    const float* gl_pn_b   = (const float*)d_in[pi++];
    const float* gl_gru_Wih = (const float*)d_in[pi++];
    const float* gl_gru_Whh = (const float*)d_in[pi++];
    const float* gl_gru_bih = (const float*)d_in[pi++];
    const float* gl_gru_bhh = (const float*)d_in[pi++];
    const float* gl_bn_g   = (const float*)d_in[pi++];
    const float* gl_bn_b   = (const float*)d_in[pi++];
    const float* g3_W1     = (const float*)d_in[pi++];
    const float* g3_b1     = (const float*)d_in[pi++];
    const float* g3_W2     = (const float*)d_in[pi++];
    const float* g3_b2     = (const float*)d_in[pi++];
    const float* g3_W3     = (const float*)d_in[pi++];
    const float* g3_b3     = (const float*)d_in[pi++];
    const float* g3_bn_g   = (const float*)d_in[pi++];
    const float* g3_bn_b   = (const float*)d_in[pi++];
    const float* ro_W      = (const float*)d_in[pi++];
    const float* ro_b      = (const float*)d_in[pi++];

    // ---- workspace carve-out ----
    size_t cur = 0;
    auto alloc = [&](size_t elems) -> float* {
        char* p = (char*)d_ws + cur;
        cur += ((elems * sizeof(float) + 255) & ~(size_t)255);
        return (float*)p;
    };
    float*    hv   = alloc((size_t)N * G);          // hv_new
    float*    X1   = alloc((size_t)E * KP1);        // padded concat input for he1
    float*    bufA = alloc((size_t)E * G);          // he1 / e3 ping
    float*    bufB = alloc((size_t)E * G);          // et  / e3 pong
    float*    ew   = alloc((size_t)E);              // logits -> exp -> attn / tanh weights
    unsigned* nmax = (unsigned*)alloc((size_t)N);
    float*    nsum = alloc((size_t)N);
    float*    ctx  = alloc((size_t)N * G);          // ctx / ctx2
    float*    gi   = alloc((size_t)N * 3 * G);
    float*    gh   = alloc((size_t)N * 3 * G);
    float*    h1   = alloc((size_t)N * G);
    float*    h2   = alloc((size_t)N * G);
    float*    hvp  = alloc((size_t)N * G);
    float*    sumf = alloc((size_t)N * G);
    float*    X3   = alloc((size_t)E * KP3);        // padded g3 input
    float*    pe1Wp = alloc((size_t)G * KP1);       // gc_pe1_W zero-padded K 50->52
    float*    g3W1p = alloc((size_t)G * KP3);       // g3_W1    zero-padded K 201->204
    float*    mu   = alloc((size_t)G);
    float*    var  = alloc((size_t)G);
    float*    gsum = alloc((size_t)NG * G);
    unsigned* gmax = (unsigned*)alloc((size_t)NG * G);

    auto blocks = [](size_t n) { return dim3((unsigned)((n + 255) / 256)); };
    auto gemm = [&](const float* A, int lda, const float* W, int ldw, const float* bias,
                    float* C, int ldc, int M, int Nn, int K, int act) {
        dim3 grid((M + 127) / 128, (Nn + 63) / 64);
        gemm_wmma_f32<<<grid, 128, 0, stream>>>(A, lda, W, ldw, bias, C, ldc, M, Nn, K, act);
    };

    // ---- one-time weight pads (K to multiple of 4 / even ld) ----
    pad_w_k<<<blocks((size_t)G * KP1), 256, 0, stream>>>(pe1Wp, gc_pe1_W, G, NF + EF, KP1);
    pad_w_k<<<blocks((size_t)G * KP3), 256, 0, stream>>>(g3W1p, g3_W1, G, G + 1, KP3);

    // ================= GetContext (AttentiveGRU1) =================
    gemm(node, NF, gc_pn_W, NF, gc_pn_b, hv, G, N, G, NF, 1);                  // hv_new
    build_X1_k<<<blocks((size_t)E * KP1), 256, 0, stream>>>(node, edgef, src, X1, E, NF, EF, KP1);
    gemm(X1, KP1, pe1Wp, KP1, gc_pe1_b, bufA, G, E, G, KP1, 1);                // he1
    gemm(bufA, G, gc_et_W, G, gc_et_b, bufB, G, E, G, G, 0);                   // et
    edge_logits_k<<<blocks(E), 256, 0, stream>>>(gc_pe2_W, gc_pe2_b, hv, dst, bufA, nullptr, ew, E, G);
    fill_u32_k<<<blocks(N), 256, 0, stream>>>(nmax, NEG_INF_ORD, (size_t)N);
    fill_f32_k<<<blocks(N), 256, 0, stream>>>(nsum, 0.0f, (size_t)N);
    segmax_k<<<blocks(E), 256, 0, stream>>>(ew, dst, nmax, E);
    seg_expsum_k<<<blocks(E), 256, 0, stream>>>(ew, dst, nmax, nsum, E);
    fill_f32_k<<<blocks((size_t)N * G), 256, 0, stream>>>(ctx, 0.0f, (size_t)N * G);
    scatter_attn_k<<<blocks((size_t)E * G), 256, 0, stream>>>(ew, nsum, dst, nullptr, bufB, ctx,
                                                              (size_t)E * G, G, 0.0f);
    elu_ip_k<<<blocks((size_t)N * G), 256, 0, stream>>>(ctx, (size_t)N * G);
    gemm(ctx, G, gc_gru_Wih, G, gc_gru_bih, gi, 3 * G, N, 3 * G, G, 0);
    gemm(hv,  G, gc_gru_Whh, G, gc_gru_bhh, gh, 3 * G, N, 3 * G, G, 0);
    gru_relu_k<<<blocks((size_t)N * G), 256, 0, stream>>>(gi, gh, hv, h1, N, G);   // h

    // ================= GNNLayer (AttentiveGRU2) =================
    edge_logits_k<<<blocks(E), 256, 0, stream>>>(gl_pe_W, gl_pe_b, h1, dst, h1, src, ew, E, G);
    fill_u32_k<<<blocks(N), 256, 0, stream>>>(nmax, NEG_INF_ORD, (size_t)N);
    fill_f32_k<<<blocks(N), 256, 0, stream>>>(nsum, 0.0f, (size_t)N);
    segmax_k<<<blocks(E), 256, 0, stream>>>(ew, dst, nmax, E);
    seg_expsum_k<<<blocks(E), 256, 0, stream>>>(ew, dst, nmax, nsum, E);
    gemm(h1, G, gl_pn_W, G, gl_pn_b, hvp, G, N, G, G, 0);                      // hvp
    fill_f32_k<<<blocks((size_t)N * G), 256, 0, stream>>>(ctx, 0.0f, (size_t)N * G);
    scatter_attn_k<<<blocks((size_t)E * G), 256, 0, stream>>>(ew, nsum, dst, src, hvp, ctx,
                                                              (size_t)E * G, G, 1.0f);
    elu_ip_k<<<blocks((size_t)N * G), 256, 0, stream>>>(ctx, (size_t)N * G);   // ctx2
    gemm(ctx, G, gl_gru_Wih, G, gl_gru_bih, gi, 3 * G, N, 3 * G, G, 0);
    gemm(h1,  G, gl_gru_Whh, G, gl_gru_bhh, gh, 3 * G, N, 3 * G, G, 0);
    gru_relu_k<<<blocks((size_t)N * G), 256, 0, stream>>>(gi, gh, h1, h2, N, G);
    colstats_k<<<G, 256, 0, stream>>>(h2, N, G, mu, var);
    bn_apply_k<<<blocks((size_t)N * G), 256, 0, stream>>>(h2, h1, mu, var, gl_bn_g, gl_bn_b,
                                                          sumf, (size_t)N * G, G);  // sum_feats

    // ================= DTIConvGraph3Layer =================
    build_X3_k<<<blocks((size_t)E * KP3), 256, 0, stream>>>(bond, sumf, src, dst, X3, E, G, KP3);
    gemm(X3, KP3, g3W1p, KP3, g3_b1, bufA, G, E, G, KP3, 1);
    gemm(bufA, G, g3_W2, G, g3_b2, bufB, G, E, G, G, 1);
    gemm(bufB, G, g3_W3, G, g3_b3, bufA, G, E, G, G, 1);
    colstats_k<<<G, 256, 0, stream>>>(bufA, E, G, mu, var);
    bn_apply_k<<<blocks((size_t)E * G), 256, 0, stream>>>(bufA, nullptr, mu, var, g3_bn_g, g3_bn_b,
                                                          bufB, (size_t)E * G, G);   // e3

    // ================= EdgeWeightedSumAndMax readout =================
    rowdot_k<<<blocks(E), 256, 0, stream>>>(bufB, G, ro_W, ro_b, ew, E, G, 2);       // tanh weights
    fill_f32_k<<<blocks((size_t)NG * G), 256, 0, stream>>>(gsum, 0.0f, (size_t)NG * G);
    fill_u32_k<<<blocks((size_t)NG * G), 256, 0, stream>>>(gmax, NEG_INF_ORD, (size_t)NG * G);
    readout_scatter_k<<<blocks((size_t)E * G), 256, 0, stream>>>(bufB, ew, egid, gsum, gmax, E, G);
    pack_out_k<<<blocks((size_t)NG * G), 256, 0, stream>>>(gsum, gmax, (float*)d_out, NG, G);
}